// LinearTransfer1_1_simple_74706661146736
// MI455X (gfx1250) — compile-verified
//
#include <hip/hip_runtime.h>

// ---------------------------------------------------------------------------
// LinearTransfer graph op for MI455X (gfx1250, wave32).
// Bandwidth/atomic-bound scatter-gather: no matrix ops apply. Strategy:
//  - b128 (float4) row accesses: one wave32 covers a 128-float channel row.
//  - CSR offsets via binary search over the *sorted* indicator arrays.
//  - Fuse x_int production with its two consumers (never materialize 256 MB).
//  - ~230 MB working set vs 192 MB L2 -> gathers mostly hit L2.
//  - global_prefetch_b8 on index-chased rows; scatters pinned to the
//    non-returning L2 atomic global_atomic_add_f32 via inline asm (avoids any
//    possible CAS-loop fallback from the generic atomic lowering).
// ---------------------------------------------------------------------------

#define NCV 32  // 128 channels = 32 float4

__device__ __forceinline__ void atom_add(float* p, float v) {
#if defined(__HIP_DEVICE_COMPILE__) && defined(__gfx1250__)
  // Non-returning L2-side f32 atomic add (VGLOBAL opcode 86). STOREcnt-tracked;
  // the implicit wait-idle at s_endpgm covers completion, and pass ordering is
  // enforced by kernel launch boundaries on the stream.
  asm volatile("global_atomic_add_f32 %0, %1, off" : : "v"(p), "v"(v) : "memory");
#else
  __hip_atomic_fetch_add(p, v, __ATOMIC_RELAXED, __HIP_MEMORY_SCOPE_AGENT);
#endif
}

__device__ __forceinline__ int lower_bound_i(const int* __restrict__ a, int n, int key) {
  int lo = 0, hi = n;
  while (lo < hi) {
    int mid = (lo + hi) >> 1;
    if (a[mid] < key) lo = mid + 1; else hi = mid;
  }
  return lo;
}

// off[i] = first index in sorted[] with value >= i, for i in [0, nsegs]
__global__ void seg_offsets_kernel(const int* __restrict__ sorted, int n,
                                   int* __restrict__ off, int nsegs) {
  int i = blockIdx.x * blockDim.x + threadIdx.x;
  if (i <= nsegs) off[i] = lower_bound_i(sorted, n, i);
}

// x_inv[d] = mean over contiguous node rows [dom_off[d], dom_off[d+1])
__global__ void domain_mean_kernel(const float4* __restrict__ x,
                                   const int* __restrict__ dom_off,
                                   float4* __restrict__ x_inv, int num_domains) {
  int wid  = (blockIdx.x * blockDim.x + threadIdx.x) >> 5;
  int lane = threadIdx.x & 31;
  if (wid >= num_domains) return;
  int s = dom_off[wid], e = dom_off[wid + 1];
  float4 acc = make_float4(0.f, 0.f, 0.f, 0.f);
  for (int r = s; r < e; ++r) {
    float4 v = x[(size_t)r * NCV + lane];
    acc.x += v.x; acc.y += v.y; acc.z += v.z; acc.w += v.w;
  }
  int cnt = e - s;
  float inv = 1.0f / (float)(cnt > 0 ? cnt : 1);
  acc.x *= inv; acc.y *= inv; acc.z *= inv; acc.w *= inv;
  x_inv[(size_t)wid * NCV + lane] = acc;
}

// dom2dom[dmap_tgt[j]] += x_inv[dmap_src[j]]   (one wave per domain edge)
__global__ void dom2dom_kernel(const float4* __restrict__ x_inv,
                               const int* __restrict__ dmap_src,
                               const int* __restrict__ dmap_tgt,
                               float* __restrict__ dom2dom, int e_dom) {
  int wid  = (blockIdx.x * blockDim.x + threadIdx.x) >> 5;
  int lane = threadIdx.x & 31;
  if (wid >= e_dom) return;
  int s = dmap_src[wid], t = dmap_tgt[wid];
  float4 v = x_inv[(size_t)s * NCV + lane];
  float* p = dom2dom + (size_t)t * 128 + lane * 4;
  atom_add(p + 0, v.x);
  atom_add(p + 1, v.y);
  atom_add(p + 2, v.z);
  atom_add(p + 3, v.w);
}

// One wave per intersection i:
//   xint = sum over edges e in segment of x[src[e]]
//   out[tgt[e]]          += e1_0 * x[src[e]] + e1_1 * xint   (fused y_local+y_bint)
//   invint[dmap_tgt[i]]  += e0_0 * xint                      (pre-scaled inv_int)
__global__ void intersect_kernel(const float4* __restrict__ x,
                                 const int* __restrict__ int_off,
                                 const int* __restrict__ np_src,
                                 const int* __restrict__ np_tgt,
                                 const int* __restrict__ dmap_tgt,
                                 const float* __restrict__ eps1,
                                 const float* __restrict__ eps0,
                                 float* __restrict__ out,
                                 float* __restrict__ invint, int n_int) {
  int wid  = (blockIdx.x * blockDim.x + threadIdx.x) >> 5;
  int lane = threadIdx.x & 31;
  if (wid >= n_int) return;
  int s = int_off[wid], e = int_off[wid + 1];
  if (s == e) return;  // empty intersection contributes zeros everywhere

  float e1_0 = eps1[0], e1_1 = eps1[1], e0_0 = eps0[0];

  float4 xint = make_float4(0.f, 0.f, 0.f, 0.f);
  for (int k = s; k < e; ++k) {
    if (k + 1 < e) {
      // hide the index-chased gather latency (global_prefetch_b8)
      __builtin_prefetch(&x[(size_t)np_src[k + 1] * NCV + lane], 0, 3);
    }
    float4 v = x[(size_t)np_src[k] * NCV + lane];
    xint.x += v.x; xint.y += v.y; xint.z += v.z; xint.w += v.w;
  }

  // scatter fused (e1_0 * x_src + e1_1 * xint) per edge; rows hot in L0/L2
  for (int k = s; k < e; ++k) {
    int r = np_src[k], t = np_tgt[k];
    float4 v = x[(size_t)r * NCV + lane];
    float* p = out + (size_t)t * 128 + lane * 4;
    atom_add(p + 0, e1_0 * v.x + e1_1 * xint.x);
    atom_add(p + 1, e1_0 * v.y + e1_1 * xint.y);
    atom_add(p + 2, e1_0 * v.z + e1_1 * xint.z);
    atom_add(p + 3, e1_0 * v.w + e1_1 * xint.w);
  }

  int dt = dmap_tgt[wid];
  float* q = invint + (size_t)dt * 128 + lane * 4;
  atom_add(q + 0, e0_0 * xint.x);
  atom_add(q + 1, e0_0 * xint.y);
  atom_add(q + 2, e0_0 * xint.z);
  atom_add(q + 3, e0_0 * xint.w);
}

// out[n] += e1_3*x[n] + invint[d] + (e1_2 + e0_1)*dom2dom[d] + e0_2*x_inv[d]
__global__ void finalize_kernel(const float4* __restrict__ x,
                                const int* __restrict__ dom_ind,
                                const float4* __restrict__ x_inv,
                                const float4* __restrict__ dom2dom,
                                const float4* __restrict__ invint,
                                const float* __restrict__ eps1,
                                const float* __restrict__ eps0,
                                float4* __restrict__ out, int n_nodes) {
  int wid  = (blockIdx.x * blockDim.x + threadIdx.x) >> 5;
  int lane = threadIdx.x & 31;
  if (wid >= n_nodes) return;
  int d = dom_ind[wid];
  float e1_3 = eps1[3];
  float c_dd = eps1[2] + eps0[1];  // y_bdom + rep0 dom2dom terms share the gather
  float c_xi = eps0[2];

  size_t no   = (size_t)wid * NCV + lane;
  size_t dofs = (size_t)d * NCV + lane;
  float4 o  = out[no];
  float4 xv = x[no];
  float4 dd = dom2dom[dofs];
  float4 ii = invint[dofs];   // already scaled by e0_0
  float4 xi = x_inv[dofs];
  o.x += e1_3 * xv.x + ii.x + c_dd * dd.x + c_xi * xi.x;
  o.y += e1_3 * xv.y + ii.y + c_dd * dd.y + c_xi * xi.y;
  o.z += e1_3 * xv.z + ii.z + c_dd * dd.z + c_xi * xi.z;
  o.w += e1_3 * xv.w + ii.w + c_dd * dd.w + c_xi * xi.w;
  out[no] = o;
}

extern "C" void kernel_launch(void* const* d_in, const int* in_sizes, int n_in,
                              void* d_out, int out_size, void* d_ws, size_t ws_size,
                              hipStream_t stream) {
  const float* x        = (const float*)d_in[0];
  const float* eps1     = (const float*)d_in[1];
  const float* eps0     = (const float*)d_in[2];
  const int*   dom_ind  = (const int*)d_in[3];
  const int*   np_src   = (const int*)d_in[4];
  const int*   np_tgt   = (const int*)d_in[5];
  const int*   int_ind  = (const int*)d_in[6];
  const int*   dmap_src = (const int*)d_in[7];
  const int*   dmap_tgt = (const int*)d_in[8];
  float*       out      = (float*)d_out;

  const int N      = in_sizes[0] / 128;  // 150000
  const int E_NODE = in_sizes[4];        // 1000000
  const int E_DOM  = in_sizes[7];        // 500000
  const int D      = 50000;              // num_domains (device scalar; fixed by reference)
  const int N_INT  = E_DOM;              // n_int = dmap_src.shape[0]

  // workspace layout
  char* ws = (char*)d_ws;
  size_t off = 0;
  auto carve = [&](size_t bytes) -> void* {
    void* p = ws + off;
    off = (off + bytes + 255) & ~(size_t)255;
    return p;
  };
  float* x_inv   = (float*)carve((size_t)D * 128 * sizeof(float));
  float* dom2dom = (float*)carve((size_t)D * 128 * sizeof(float));
  float* invint  = (float*)carve((size_t)D * 128 * sizeof(float));
  int*   dom_off = (int*)carve((size_t)(D + 1) * sizeof(int));
  int*   int_off = (int*)carve((size_t)(N_INT + 1) * sizeof(int));

  // zero the accumulators every call (graph-replay safe)
  hipMemsetAsync(dom2dom, 0, (size_t)D * 128 * sizeof(float), stream);
  hipMemsetAsync(invint,  0, (size_t)D * 128 * sizeof(float), stream);
  hipMemsetAsync(out,     0, (size_t)N * 128 * sizeof(float), stream);

  // CSR offsets from sorted indicators
  {
    int n = D + 1;
    seg_offsets_kernel<<<(n + 255) / 256, 256, 0, stream>>>(dom_ind, N, dom_off, D);
  }
  {
    int n = N_INT + 1;
    seg_offsets_kernel<<<(n + 255) / 256, 256, 0, stream>>>(int_ind, E_NODE, int_off, N_INT);
  }

  // Pass A: per-domain mean
  {
    long long thr = (long long)D * 32;
    domain_mean_kernel<<<(int)((thr + 255) / 256), 256, 0, stream>>>(
        (const float4*)x, dom_off, (float4*)x_inv, D);
  }
  // Pass B: domain->domain transfer
  {
    long long thr = (long long)E_DOM * 32;
    dom2dom_kernel<<<(int)((thr + 255) / 256), 256, 0, stream>>>(
        (const float4*)x_inv, dmap_src, dmap_tgt, dom2dom, E_DOM);
  }
  // Pass C: fused intersection reduce + y_local/y_bint scatter + inv_int
  {
    long long thr = (long long)N_INT * 32;
    intersect_kernel<<<(int)((thr + 255) / 256), 256, 0, stream>>>(
        (const float4*)x, int_off, np_src, np_tgt, dmap_tgt,
        eps1, eps0, out, invint, N_INT);
  }
  // Pass D: per-node epsilon combine
  {
    long long thr = (long long)N * 32;
    finalize_kernel<<<(int)((thr + 255) / 256), 256, 0, stream>>>(
        (const float4*)x, dom_ind, (const float4*)x_inv, (const float4*)dom2dom,
        (const float4*)invint, eps1, eps0, (float4*)out, N);
  }
}